// STG_NF_62551903699503
// MI455X (gfx1250) — compile-verified
//
#include <hip/hip_runtime.h>

typedef __attribute__((ext_vector_type(2))) float v2f;
typedef __attribute__((ext_vector_type(8))) float v8f;

#define TT   24
#define VVV  18
#define KK   8
#define TKK  13
#define PADK 6
#define SPB  16            // samples per block (= WMMA N)
#define NTH  (SPB * TT)    // 384 threads = 12 waves

__launch_bounds__(NTH)
__global__ void stgnf_flow_kernel(
    const float* __restrict__ x,
    const float* __restrict__ an_bias, const float* __restrict__ an_logs,
    const float* __restrict__ gcn_w,   const float* __restrict__ gcn_b,
    const float* __restrict__ bn1,     const float* __restrict__ tcn_w,
    const float* __restrict__ tcn_b,   const float* __restrict__ bn2,
    const float* __restrict__ res_w,   const float* __restrict__ res_bn,
    float* __restrict__ out, int nsamp)
{
    __shared__ float lds_W[48 * 48];   // banded conv matrix, rows=(co,t), cols=(ci,t')
    __shared__ float lds_h[48 * SPB];  // conv inputs  [ (ci*24+t) ][ sample ]
    __shared__ float lds_u[48 * SPB];  // conv outputs [ (co*24+t) ][ sample ]
    __shared__ float lds_red[NTH];     // per-sample reduction

    const int tid = threadIdx.x;
    const int sl  = tid / TT;          // local sample (WMMA column)
    const int t   = tid % TT;          // time index
    const long sample = (long)blockIdx.x * SPB + sl;
    const bool active = sample < nsamp;

    // Register-resident state: zA = channel 0, zB = channel 1 (per v)
    float zA[VVV], zB[VVV];
    {
        const float* xp = x + sample * (2 * TT * VVV) + t * VVV;
        #pragma unroll
        for (int v = 0; v < VVV; ++v) zA[v] = active ? xp[v] : 0.f;
        #pragma unroll
        for (int v = 0; v < VVV; ++v) zB[v] = active ? xp[TT * VVV + v] : 0.f;
    }

    float acc_logscale = 0.f;          // sum_k sum_v log(scale)

    for (int k = 0; k < KK; ++k) {
        // ---- build banded conv weight matrix for this flow step ----
        const float* w = tcn_w + k * (2 * 2 * TKK);
        for (int e = tid; e < 48 * 48; e += NTH) {
            int mrow = e / 48, jcol = e % 48;
            int co = mrow / TT, to = mrow % TT;
            int ci = jcol / TT, ti = jcol % TT;
            int tap = ti - to + PADK;
            float val = 0.f;
            if (tap >= 0 && tap < TKK) val = w[(co * 2 + ci) * TKK + tap];
            lds_W[e] = val;
        }

        // ---- per-channel affine + channel flip (in registers) ----
        const float b0 = an_bias[k * 2 + 0], b1 = an_bias[k * 2 + 1];
        const float e0 = __expf(an_logs[k * 2 + 0]);
        const float e1 = __expf(an_logs[k * 2 + 1]);
        #pragma unroll
        for (int v = 0; v < VVV; ++v) {
            float a_old = zA[v];
            zA[v] = (zB[v] + b1) * e1;     // z1
            zB[v] = (a_old + b0) * e0;     // z2 (pre-update)
        }

        // ---- mean over V (thread-local), gcn affine, bn1, relu ----
        float m = 0.f;
        #pragma unroll
        for (int v = 0; v < VVV; ++v) m += zA[v];
        m *= (1.0f / VVV);
        #pragma unroll
        for (int c = 0; c < 2; ++c) {
            float hv  = m * gcn_w[k * 2 + c] + gcn_b[k * 2 + c];
            float g   = bn1[k * 8 + 0 * 2 + c], bb = bn1[k * 8 + 1 * 2 + c];
            float mu  = bn1[k * 8 + 2 * 2 + c], va = bn1[k * 8 + 3 * 2 + c];
            hv = (hv - mu) * (g * rsqrtf(va + 1e-5f)) + bb;
            lds_h[(c * TT + t) * SPB + sl] = fmaxf(hv, 0.f);
        }
        __syncthreads();

        // ---- temporal conv as WMMA: D[48x16] = W[48x48] * H[48x16] ----
        if (tid < 32) {                   // one full wave, EXEC all ones
            const int lane = tid;
            const int col  = lane & 15;   // sample column N
            const int half = lane >> 4;   // K sub-select for A/B frags
            const int mr   = lane & 15;   // M row within tile
            #pragma unroll
            for (int mt = 0; mt < 3; ++mt) {
                v8f acc = {0.f, 0.f, 0.f, 0.f, 0.f, 0.f, 0.f, 0.f};
                const int mrow = mt * 16 + mr;
                #pragma unroll
                for (int k4 = 0; k4 < 12; ++k4) {
                    const int kb = k4 * 4 + half * 2;
                    v2f a, b;
                    a.x = lds_W[mrow * 48 + kb];
                    a.y = lds_W[mrow * 48 + kb + 1];
                    b.x = lds_h[(kb    ) * SPB + col];
                    b.y = lds_h[(kb + 1) * SPB + col];
                    acc = __builtin_amdgcn_wmma_f32_16x16x4_f32(
                        false, a, false, b, (short)0, acc, false, false);
                }
                #pragma unroll
                for (int r = 0; r < 8; ++r)
                    lds_u[(mt * 16 + r + 8 * half) * SPB + col] = acc[r];
            }
        }
        __syncthreads();

        // ---- read conv result, + bias, bn2 ----
        float uu[2];
        uu[0] = lds_u[(0 * TT + t) * SPB + sl];
        uu[1] = lds_u[(1 * TT + t) * SPB + sl];
        #pragma unroll
        for (int c = 0; c < 2; ++c) {
            float g  = bn2[k * 8 + 0 * 2 + c], bb = bn2[k * 8 + 1 * 2 + c];
            float mu = bn2[k * 8 + 2 * 2 + c], va = bn2[k * 8 + 3 * 2 + c];
            uu[c] = (uu[c] + tcn_b[k * 2 + c] - mu) * (g * rsqrtf(va + 1e-5f)) + bb;
        }

        // ---- residual bn as affine in z1 (k>0) ----
        float rcoef[2] = {0.f, 0.f}, rcon[2] = {0.f, 0.f};
        if (k > 0) {
            #pragma unroll
            for (int c = 0; c < 2; ++c) {
                float g  = res_bn[k * 8 + 0 * 2 + c], bb = res_bn[k * 8 + 1 * 2 + c];
                float mu = res_bn[k * 8 + 2 * 2 + c], va = res_bn[k * 8 + 3 * 2 + c];
                float s  = g * rsqrtf(va + 1e-5f);
                rcoef[c] = res_w[k * 2 + c] * s;
                rcon[c]  = bb - mu * s;
            }
        }

        // ---- coupling: shift/scale per element + log-det ----
        float lsum = 0.f;
        #pragma unroll
        for (int v = 0; v < VVV; ++v) {
            float h0 = uu[0], h1 = uu[1];
            if (k > 0) {
                h0 += rcoef[0] * zA[v] + rcon[0];
                h1 += rcoef[1] * zA[v] + rcon[1];
            }
            float shift = fmaxf(h0, 0.f);
            float sarg  = fmaxf(h1, 0.f) + 2.0f;
            float scale = 1.0f / (1.0f + __expf(-sarg)) + 1e-6f;
            zB[v] = (zB[v] + shift) * scale;
            lsum += __logf(scale);
        }
        acc_logscale += lsum;
        // state for next step is already in zA (=z1), zB (=updated z2)
    }

    // ---- likelihood partial + per-sample reduction ----
    float q = 0.f;
    #pragma unroll
    for (int v = 0; v < VVV; ++v) q += zA[v] * zA[v] + zB[v] * zB[v];
    const float LOG2PI_ = 1.83787706640934548356f;
    lds_red[sl * TT + t] = acc_logscale - 0.5f * q - 0.5f * LOG2PI_ * (2 * VVV);
    __syncthreads();

    if (t == 0 && active) {
        float tot = 0.f;
        #pragma unroll
        for (int i = 0; i < TT; ++i) tot += lds_red[sl * TT + i];
        float cl = 0.f;
        for (int k = 0; k < KK; ++k) cl += an_logs[k * 2 + 0] + an_logs[k * 2 + 1];
        cl *= (float)(TT * VVV);
        const float LOG2_ = 0.69314718055994530942f;
        out[sample] = -(cl + tot) / (LOG2_ * 2.0f * TT * VVV);
    }
}

extern "C" void kernel_launch(void* const* d_in, const int* in_sizes, int n_in,
                              void* d_out, int out_size, void* d_ws, size_t ws_size,
                              hipStream_t stream) {
    const float* x       = (const float*)d_in[0];
    const float* an_bias = (const float*)d_in[1];
    const float* an_logs = (const float*)d_in[2];
    const float* gcn_w   = (const float*)d_in[3];
    const float* gcn_b   = (const float*)d_in[4];
    const float* bn1     = (const float*)d_in[5];
    const float* tcn_w   = (const float*)d_in[6];
    const float* tcn_b   = (const float*)d_in[7];
    const float* bn2     = (const float*)d_in[8];
    const float* res_w   = (const float*)d_in[9];
    const float* res_bn  = (const float*)d_in[10];
    float* out = (float*)d_out;

    const int nsamp  = in_sizes[0] / (2 * TT * VVV);
    const int blocks = (nsamp + SPB - 1) / SPB;
    stgnf_flow_kernel<<<blocks, NTH, 0, stream>>>(
        x, an_bias, an_logs, gcn_w, gcn_b, bn1, tcn_w, tcn_b, bn2,
        res_w, res_bn, out, nsamp);
}